// UnEmbedder_39178691674888
// MI455X (gfx1250) — compile-verified
//
#include <hip/hip_runtime.h>
#include <stdint.h>

#define SEQ   2048
#define DIM   1024
#define VOCAB 50257

#define KC        32            // K chunk (columns per TDM tile)
#define NCHUNK    (DIM / KC)    // 32 chunks (even -> 2x unrolled buffer ping-pong)
#define LDS_ROWSTRIDE 34        // 32 data DWORDs + 2 pad DWORDs (bank-conflict-free)
#define TILE_LDS_DW  (128 * LDS_ROWSTRIDE)        // dwords per staged tile
#define SMEM_DWORDS  (4 * TILE_LDS_DW)            // Y0,W0,Y1,W1

// Static LDS byte offsets for the 4 staged tiles
#define Y_OFF0  0u
#define W_OFF0  (1u * TILE_LDS_DW * 4u)
#define Y_OFF1  (2u * TILE_LDS_DW * 4u)
#define W_OFF1  (3u * TILE_LDS_DW * 4u)

typedef __attribute__((ext_vector_type(2))) float v2f;
typedef __attribute__((ext_vector_type(8))) float v8f;
typedef __attribute__((ext_vector_type(4))) unsigned int v4u;
typedef __attribute__((ext_vector_type(4))) int v4i;
typedef __attribute__((ext_vector_type(8))) int v8i;

// ---------------------------------------------------------------------------
// Stage 1: y[s,d] = (emb[s,d] - ln_bias[d]) / (ln_weight[d] + 1e-6) - pos[s,d]
// ---------------------------------------------------------------------------
__global__ __launch_bounds__(256) void prep_y(const float* __restrict__ emb,
                                              const float* __restrict__ lnw,
                                              const float* __restrict__ lnb,
                                              const float* __restrict__ pos,
                                              float* __restrict__ y) {
    int i = blockIdx.x * blockDim.x + threadIdx.x;   // over SEQ*DIM
    int d = i & (DIM - 1);
    y[i] = (emb[i] - lnb[d]) / (lnw[d] + 1e-6f) - pos[i];
}

// ---------------------------------------------------------------------------
// Stage 2: hn[v] = 0.5 * ||vocab[v]||^2   (one wave32 per vocab row)
// ---------------------------------------------------------------------------
__global__ __launch_bounds__(256) void vocab_norms(const float* __restrict__ W,
                                                   float* __restrict__ hn) {
    int wave = (int)((blockIdx.x * blockDim.x + threadIdx.x) >> 5);
    int lane = threadIdx.x & 31;
    if (wave >= VOCAB) return;
    const float* row = W + (size_t)wave * DIM;
    float s = 0.0f;
    #pragma unroll 4
    for (int k = lane; k < DIM; k += 32) { float v = row[k]; s += v * v; }
    #pragma unroll
    for (int m = 16; m >= 1; m >>= 1) s += __shfl_xor(s, m, 32);
    if (lane == 0) hn[wave] = 0.5f * s;
}

// ---------------------------------------------------------------------------
// Stage 3: reset packed (key<<32 | index) minima
// ---------------------------------------------------------------------------
__global__ __launch_bounds__(256) void init_best(unsigned long long* __restrict__ best) {
    int i = blockIdx.x * blockDim.x + threadIdx.x;
    if (i < SEQ) best[i] = 0xFFFFFFFFFFFFFFFFull;
}

// order-preserving float -> u32 map (ascending floats -> ascending uints)
__device__ __forceinline__ uint32_t fkey(float f) {
    uint32_t b = __float_as_uint(f);
    return (b & 0x80000000u) ? ~b : (b | 0x80000000u);
}

// ---------------------------------------------------------------------------
// TDM: DMA a 2D tile (tile_rows x KC f32) from global to LDS with row padding.
// D# per CDNA5 ISA. data_size=4B, pad 2 DWORDs
// every 32 DWORDs -> LDS row stride 34 DWORDs. OOB rows (beyond rem_rows)
// read as zero (handles the vocab tail).
// ---------------------------------------------------------------------------
__device__ __forceinline__ void tdm_load_2d(uint32_t lds_byte_off,
                                            const float* gaddr,
                                            uint32_t rem_rows, uint32_t rem_cols,
                                            uint32_t tile_rows, uint32_t tile_cols,
                                            uint32_t row_stride_elems) {
    uint64_t ga = (uint64_t)(uintptr_t)gaddr;
    v4u g0;
    g0[0] = 1u;                                            // count=1 (valid), user mode
    g0[1] = lds_byte_off;                                  // lds_addr
    g0[2] = (uint32_t)ga;                                  // global_addr[31:0]
    g0[3] = (uint32_t)((ga >> 32) & 0x01FFFFFFu) | (2u << 30);  // addr[56:32] | type=2

    v8i g1;
    g1[0] = (int)((2u << 16)      // data_size = 4 bytes
                | (1u << 20)      // pad_enable
                | (4u << 22)      // pad_interval code 4 -> every 32 DWORDs
                | (1u << 25));    // pad_amount  code 1 -> 2 DWORDs
    g1[1] = (int)((rem_cols & 0xFFFFu) << 16);                       // tensor_dim0[15:0]
    g1[2] = (int)((rem_cols >> 16) | ((rem_rows & 0xFFFFu) << 16));  // dim0 hi | dim1 lo
    g1[3] = (int)(((rem_rows >> 16) & 0xFFFFu) | (tile_cols << 16)); // dim1 hi | tile_dim0
    g1[4] = (int)(tile_rows & 0xFFFFu);                              // tile_dim1; tile_dim2=0
    g1[5] = (int)row_stride_elems;                                   // tensor_dim0_stride lo
    g1[6] = 0;                                                       // stride0 hi | stride1 lo
    g1[7] = 0;                                                       // stride1 hi (2D: unused)

    v4i z4 = {0, 0, 0, 0};
#if __clang_major__ >= 23
    v8i z8 = {0, 0, 0, 0, 0, 0, 0, 0};
    __builtin_amdgcn_tensor_load_to_lds(g0, g1, z4, z4, z8, 0);
#else
    __builtin_amdgcn_tensor_load_to_lds(g0, g1, z4, z4, 0);
#endif
}

// One K-chunk of WMMA work on a staged buffer (all offsets compile-time per buffer)
__device__ __forceinline__ void compute_chunk(const float* __restrict__ ybuf,
                                              const float* __restrict__ wbuf,
                                              const uint32_t aRow[2], const uint32_t bRow[4],
                                              v8f acc[2][4]) {
    #pragma unroll
    for (int kk = 0; kk < KC; kk += 4) {
        v2f a[2], b[4];
        #pragma unroll
        for (int mt = 0; mt < 2; ++mt)
            a[mt] = *(const v2f*)(ybuf + aRow[mt] + kk);
        #pragma unroll
        for (int nt = 0; nt < 4; ++nt)
            b[nt] = *(const v2f*)(wbuf + bRow[nt] + kk);

        #pragma unroll
        for (int mt = 0; mt < 2; ++mt)
            #pragma unroll
            for (int nt = 0; nt < 4; ++nt)
                acc[mt][nt] = __builtin_amdgcn_wmma_f32_16x16x4_f32(
                    false, a[mt], false, b[nt],
                    (short)0, acc[mt][nt], false, false);
    }
}

// ---------------------------------------------------------------------------
// Stage 4: LDS-staged f32 WMMA GEMM + argmin.
// Block = 256 threads (8 waves), tile 128(M) x 128(N), K chunked by 32.
// Wave (wm=wid>>1 in 0..3, wn=wid&1 in 0..1) computes 32(M) x 64(N) = 8 C tiles.
// TDM double-buffers Y/W chunk tiles in LDS (ping-pong fully unrolled so all
// LDS offsets are immediates); fragments read via ds_load_2addr_b64.
// ---------------------------------------------------------------------------
__global__ __launch_bounds__(256) void gemm_argmin(const float* __restrict__ Y,
                                                   const float* __restrict__ W,
                                                   const float* __restrict__ hn,
                                                   unsigned long long* __restrict__ best) {
    extern __shared__ float smem[];   // 4 staged tiles, 68 KB

    const int lane   = threadIdx.x & 31;
    const int wid    = threadIdx.x >> 5;
    const int wm     = wid >> 1;            // 0..3
    const int wn     = wid & 1;             // 0..1
    const int lane16 = lane & 15;
    const int khalf  = (lane >> 4) << 1;    // 0 or 2 (f32 fragment k-subcolumn)

    const int M0 = blockIdx.y * 128;        // block seq rows (SEQ divides evenly)
    const int N0 = blockIdx.x * 128;        // block vocab rows (tail zero-filled by TDM)

    const uint32_t remN = (uint32_t)(VOCAB - N0);
    const float* Yg = Y + (size_t)M0 * DIM;
    const float* Wg = W + (size_t)N0 * DIM;

    // Prologue: issue chunk 0 into buffer 0 (wave 0 only; TDM ignores EXEC,
    // descriptors are wave-uniform scalars)
    if (wid == 0) {
        tdm_load_2d(Y_OFF0, Yg, SEQ - M0, DIM, 128, KC, DIM);
        tdm_load_2d(W_OFF0, Wg, remN,     DIM, 128, KC, DIM);
    }

    v8f acc[2][4];
    #pragma unroll
    for (int mt = 0; mt < 2; ++mt)
        #pragma unroll
        for (int nt = 0; nt < 4; ++nt)
            acc[mt][nt] = (v8f){0.f, 0.f, 0.f, 0.f, 0.f, 0.f, 0.f, 0.f};

    // Per-lane LDS fragment dword offsets (row * 34 + khalf), rows fixed per wave
    uint32_t aRow[2], bRow[4];
    #pragma unroll
    for (int mt = 0; mt < 2; ++mt)
        aRow[mt] = (uint32_t)(wm * 32 + mt * 16 + lane16) * LDS_ROWSTRIDE + khalf;
    #pragma unroll
    for (int nt = 0; nt < 4; ++nt)
        bRow[nt] = (uint32_t)(wn * 64 + nt * 16 + lane16) * LDS_ROWSTRIDE + khalf;

    const float* yb0 = smem + (Y_OFF0 >> 2);
    const float* wb0 = smem + (W_OFF0 >> 2);
    const float* yb1 = smem + (Y_OFF1 >> 2);
    const float* wb1 = smem + (W_OFF1 >> 2);

    // Ping-pong over chunk pairs: even chunk uses buffer 0, odd uses buffer 1.
    for (int cc = 0; cc < NCHUNK; cc += 2) {
        // ---- even chunk (buffer 0); prefetch chunk cc+1 into buffer 1 ----
        if (wid == 0) {
            const int k1 = (cc + 1) * KC;
            tdm_load_2d(Y_OFF1, Yg + k1, SEQ - M0, DIM - k1, 128, KC, DIM);
            tdm_load_2d(W_OFF1, Wg + k1, remN,     DIM - k1, 128, KC, DIM);
            __builtin_amdgcn_s_wait_tensorcnt(2);   // buffer-0 pair complete
        }
        __syncthreads();
        compute_chunk(yb0, wb0, aRow, bRow, acc);
        __syncthreads();

        // ---- odd chunk (buffer 1); prefetch chunk cc+2 into buffer 0 ----
        if (wid == 0) {
            if (cc + 2 < NCHUNK) {
                const int k2 = (cc + 2) * KC;
                tdm_load_2d(Y_OFF0, Yg + k2, SEQ - M0, DIM - k2, 128, KC, DIM);
                tdm_load_2d(W_OFF0, Wg + k2, remN,     DIM - k2, 128, KC, DIM);
                __builtin_amdgcn_s_wait_tensorcnt(2);
            } else {
                __builtin_amdgcn_s_wait_tensorcnt(0);
            }
        }
        __syncthreads();
        compute_chunk(yb1, wb1, aRow, bRow, acc);
        __syncthreads();
    }

    // Epilogue: per C-row argmin across this wave's 64 N columns -> global atomicMin.
    // C layout: VGPR j holds M=j (lanes 0-15) / M=8+j (lanes 16-31), N = lane&15.
    const int halfSel = lane >> 4;
    #pragma unroll
    for (int mt = 0; mt < 2; ++mt) {
        #pragma unroll
        for (int j = 0; j < 8; ++j) {
            const int srow = M0 + wm * 32 + 16 * mt + 8 * halfSel + j;
            uint32_t bestKey = 0xFFFFFFFFu;
            uint32_t bestIdx = 0xFFFFFFFFu;
            #pragma unroll
            for (int nt = 0; nt < 4; ++nt) {
                int v = N0 + wn * 64 + 16 * nt + lane16;
                int vc = (v > VOCAB - 1) ? (VOCAB - 1) : v;
                float val = hn[vc] - acc[mt][nt][j];    // 0.5||w||^2 - y.w
                uint32_t key = (v < VOCAB) ? fkey(val) : 0xFFFFFFFFu;
                if (key < bestKey || (key == bestKey && (uint32_t)v < bestIdx)) {
                    bestKey = key; bestIdx = (uint32_t)v;
                }
            }
            unsigned long long packed =
                ((unsigned long long)bestKey << 32) | (unsigned long long)bestIdx;
            #pragma unroll
            for (int m = 1; m <= 8; m <<= 1) {   // stays inside each 16-lane group
                unsigned long long o = __shfl_xor(packed, m, 32);
                packed = (o < packed) ? o : packed;
            }
            if (lane16 == 0) atomicMin(&best[srow], packed);
        }
    }
}

// ---------------------------------------------------------------------------
// Stage 5: unpack winning index -> d_out (float, per harness output rule)
// ---------------------------------------------------------------------------
__global__ __launch_bounds__(256) void writeout(const unsigned long long* __restrict__ best,
                                                float* __restrict__ out) {
    int i = blockIdx.x * blockDim.x + threadIdx.x;
    if (i < SEQ) out[i] = (float)(uint32_t)(best[i] & 0xFFFFFFFFull);
}

// ---------------------------------------------------------------------------
extern "C" void kernel_launch(void* const* d_in, const int* in_sizes, int n_in,
                              void* d_out, int out_size, void* d_ws, size_t ws_size,
                              hipStream_t stream) {
    const float* emb = (const float*)d_in[0];   // [SEQ, DIM]
    const float* lnw = (const float*)d_in[1];   // [DIM]
    const float* lnb = (const float*)d_in[2];   // [DIM]
    const float* pos = (const float*)d_in[3];   // [SEQ, DIM]
    const float* voc = (const float*)d_in[4];   // [VOCAB, DIM]

    char* ws = (char*)d_ws;
    float* y  = (float*)ws;                                   // 8 MiB
    float* hn = (float*)(ws + (size_t)SEQ * DIM * sizeof(float));
    unsigned long long* best =
        (unsigned long long*)(ws + (size_t)SEQ * DIM * sizeof(float)
                                 + (size_t)50432 * sizeof(float));

    prep_y<<<(SEQ * DIM) / 256, 256, 0, stream>>>(emb, lnw, lnb, pos, y);
    vocab_norms<<<(VOCAB + 7) / 8, 256, 0, stream>>>(voc, hn);
    init_best<<<(SEQ + 255) / 256, 256, 0, stream>>>(best);

    dim3 grid((VOCAB + 127) / 128, SEQ / 128);
    size_t smem_bytes = (size_t)SMEM_DWORDS * sizeof(float);   // 68 KB
    gemm_argmin<<<grid, 256, smem_bytes, stream>>>(y, voc, hn, best);

    writeout<<<(SEQ + 255) / 256, 256, 0, stream>>>(best, (float*)d_out);
}